// CrossSectionalAttentionFusionCorrelation_91225105367184
// MI455X (gfx1250) — compile-verified
//
#include <hip/hip_runtime.h>
#include <hip/hip_bf16.h>
#include <cstdint>

// CDNA5 (gfx1250) wave32 + f32 WMMA + async global->LDS implementation.
typedef __attribute__((ext_vector_type(2))) float v2f;
typedef __attribute__((ext_vector_type(8))) float v8f;

#define BDIM 32
#define CDIM 1024
#define HDIM 32
#define WDIM 32
#define RROWS (BDIM * CDIM * HDIM)     // 1,048,576 rows of length 32
#define NFLAT (RROWS * WDIM)           // 33,554,432 elements per tensor
#define CORR_BLOCKS 256

static __device__ __forceinline__ v8f wmma_f32(v2f a, v2f b, v8f c) {
    // D(16x16,f32) = A(16x4,f32) x B(4x16,f32) + C
    return __builtin_amdgcn_wmma_f32_16x16x4_f32(
        /*neg_a=*/false, a, /*neg_b=*/false, b,
        /*c_mod=*/(short)0, c, /*reuse_a=*/false, /*reuse_b=*/false);
}

// Async DMA of 16 bytes global -> LDS (tracked by ASYNCcnt, bypasses VGPRs).
// The low 32 bits of a generic pointer to __shared__ are the LDS byte offset
// (flat LDS translation uses addr[31:0]), which is what VDST expects.
static __device__ __forceinline__ void async_copy_b128(void* lds_dst, const void* gsrc) {
    const unsigned lds_off = (unsigned)(uintptr_t)lds_dst;
    asm volatile("global_load_async_to_lds_b128 %0, %1, off"
                 :: "v"(lds_off), "v"(gsrc) : "memory");
}

// ---------------------------------------------------------------------------
// Kernel 1: per-block partial Gram  P[blk][i][j] += f2[r,i] * l2[r,j]
// via 2x2 tiles of V_WMMA_F32_16X16X4_F32 (K = 4 rows per op). Fuses the
// "return lateral_features" copy. Partials go to d_out[0 .. 256*1024)
// (overwritten later by the conv kernel) so the final reduction is done in a
// FIXED order -> fully deterministic, no float atomics.
// Grid: 256 blocks x 256 threads = 2048 waves, 512 rows/wave.
// ---------------------------------------------------------------------------
__global__ __launch_bounds__(256)
void corr_kernel(const float* __restrict__ f, const float* __restrict__ l,
                 float* __restrict__ partials, float* __restrict__ lcopy) {
    __shared__ float red[8][1024];  // 32 KB: per-wave partial 32x32 matrices

    const int lane = threadIdx.x & 31;
    const int wvb  = threadIdx.x >> 5;
    const int gw   = blockIdx.x * 8 + wvb;
    const int half = lane >> 4;      // 0: lanes 0-15, 1: lanes 16-31
    const int l16  = lane & 15;
    const long rbase = (long)gw * 512;

    v8f acc00 = {}, acc01 = {}, acc10 = {}, acc11 = {};

    for (int c = 0; c < 512; c += 4) {
        // A/B fragment rows: lanes 0-15 hold K=0,1; lanes 16-31 hold K=2,3.
        const long r0 = rbase + c + (half ? 2 : 0);
        const float* frow0 = f + r0 * WDIM;
        const float* frow1 = frow0 + WDIM;
        const float* lrow0 = l + r0 * WDIM;
        const float* lrow1 = lrow0 + WDIM;
        float* crow0 = lcopy + r0 * WDIM;
        float* crow1 = crow0 + WDIM;

        float fx0 = frow0[l16],      fy0 = frow1[l16];
        float fx1 = frow0[l16 + 16], fy1 = frow1[l16 + 16];
        float lx0 = lrow0[l16],      ly0 = lrow1[l16];
        float lx1 = lrow0[l16 + 16], ly1 = lrow1[l16 + 16];

        // fused second output: lateral_features pass-through
        crow0[l16]      = lx0;  crow1[l16]      = ly0;
        crow0[l16 + 16] = lx1;  crow1[l16 + 16] = ly1;

        v2f a0, a1, b0, b1;
        a0.x = fx0 * fx0; a0.y = fy0 * fy0;   // f^2, i-tile 0 (cols 0-15)
        a1.x = fx1 * fx1; a1.y = fy1 * fy1;   // i-tile 1 (cols 16-31)
        b0.x = lx0 * lx0; b0.y = ly0 * ly0;   // l^2, j-tile 0
        b1.x = lx1 * lx1; b1.y = ly1 * ly1;   // j-tile 1

        acc00 = wmma_f32(a0, b0, acc00);
        acc01 = wmma_f32(a0, b1, acc01);
        acc10 = wmma_f32(a1, b0, acc10);
        acc11 = wmma_f32(a1, b1, acc11);
    }

    // C/D layout: VGPR v -> M = v (lanes 0-15) or v+8 (lanes 16-31), N = lane%16.
#pragma unroll
    for (int v = 0; v < 8; ++v) {
        const int iloc = v + (half ? 8 : 0);
        red[wvb][(iloc)      * 32 + l16]      = acc00[v];
        red[wvb][(iloc)      * 32 + 16 + l16] = acc01[v];
        red[wvb][(16 + iloc) * 32 + l16]      = acc10[v];
        red[wvb][(16 + iloc) * 32 + 16 + l16] = acc11[v];
    }
    __syncthreads();

    for (int idx = threadIdx.x; idx < 1024; idx += 256) {
        float s = 0.f;
#pragma unroll
        for (int w = 0; w < 8; ++w) s += red[w][idx];
        partials[blockIdx.x * 1024 + idx] = s;   // deterministic: no atomics
    }
}

// ---------------------------------------------------------------------------
// Kernel 2: fixed-order reduction of the 256 block partials, then
// A = softmax(sqrt(M), axis=-1) on the 32x32 result. One block, wave per row.
// ---------------------------------------------------------------------------
__global__ void softmax_kernel(const float* __restrict__ P, float* __restrict__ A) {
    const int j = threadIdx.x;   // column (lane)
    const int i = threadIdx.y;   // row
    float s = 0.f;
    for (int p = 0; p < CORR_BLOCKS; ++p)   // fixed order -> deterministic
        s += P[p * 1024 + i * 32 + j];
    const float m = sqrtf(s);
    float mx = m;
#pragma unroll
    for (int d = 16; d > 0; d >>= 1) mx = fmaxf(mx, __shfl_xor(mx, d, 32));
    const float e = __expf(m - mx);
    float sum = e;
#pragma unroll
    for (int d = 16; d > 0; d >>= 1) sum += __shfl_xor(sum, d, 32);
    A[i * 32 + j] = e / sum;
}

// ---------------------------------------------------------------------------
// Kernel 3: fused lateral-transfer + 1x1 conv.
//   out[b,o,h,w] = sum_k Wf[o,k] * f[b,k,h,w] + sum_k Wf[o,C+k] * lt[b,k,h,w]
//   lt[b,k,h,w]  = sum_m A[w,m] * l[b,k,h,m]      (built on-the-fly in LDS)
// Block tile: 128(M=C_out) x 128(N = 4 full h-rows) x 32(K), 8 waves as 4x2,
// each wave owns a 32x64 accumulator = 2x4 f32 WMMA tiles, K-inner step 4.
// Feature-map tiles are DMA'd with double-buffered GLOBAL_LOAD_ASYNC_TO_LDS
// (4 b128 ops/wave/K-step, s_wait_asynccnt 4 retires the previous batch) so
// HBM/L2 traffic overlaps the WMMA stream. The f-half feeds WMMA straight
// from the async buffer; only the lateral half runs the A-weighting pass.
// Grid: (8, 8, 32) = (Mblk, HWblk, batch).
// ---------------------------------------------------------------------------
__global__ __launch_bounds__(256)
void fused_conv_kernel(const float* __restrict__ f, const float* __restrict__ l,
                       const float* __restrict__ Wf, const float* __restrict__ Asm,
                       float* __restrict__ out) {
    __shared__ __align__(16) float As[128][36];       // weights; stride 36 -> conflict-free cols
    __shared__ __align__(16) float Raw[2][32][136];   // async-staged f/l rows (128 valid+pad)
    __shared__ __align__(16) float Bs[32][136];       // A-weighted lateral tile
    __shared__ float Am[32][33];                      // softmax matrix, padded

    const int tid  = threadIdx.x;
    const int lane = tid & 31;
    const int wv   = tid >> 5;
    const int wm   = wv >> 1;        // 0..3  (M sub-tile)
    const int wn   = wv & 1;         // 0..1  (N sub-tile)
    const int half = lane >> 4;
    const int l16  = lane & 15;

    const int mb = blockIdx.x;           // M block (128 rows of C_out)
    const int h0 = blockIdx.y * 4;       // four h-rows => 128 N columns
    const int b  = blockIdx.z;

    // per-(b,h0) bases; one N block spans 128 contiguous floats per channel
    const float* fB = f + (long)b * CDIM * 1024 + h0 * 32;
    const float* lB = l + (long)b * CDIM * 1024 + h0 * 32;

    // prologue: async-stage the first K tile (kk = 0, frontal half)
#pragma unroll
    for (int i = 0; i < 4; ++i) {
        const int idx = tid + i * 256;
        const int k = idx >> 5, n4 = (idx & 31) * 4;
        async_copy_b128(&Raw[0][k][n4], &fB[k * 1024 + n4]);
    }

    for (int idx = tid; idx < 1024; idx += 256)
        Am[idx >> 5][idx & 31] = Asm[idx];

    v8f acc[2][4];
#pragma unroll
    for (int i = 0; i < 2; ++i)
#pragma unroll
        for (int j = 0; j < 4; ++j) acc[i][j] = (v8f){};

    for (int kk = 0; kk < 2 * CDIM; kk += 32) {
        const int buf = (kk >> 5) & 1;

        // stage weights (synchronous float4; L2-hot, single-buffered)
        for (int idx = tid; idx < 1024; idx += 256) {
            const int m = idx >> 3, k4i = (idx & 7) * 4;
            const float4 w4 = *(const float4*)&Wf[(long)(mb * 128 + m) * (2 * CDIM) + kk + k4i];
            *(float4*)&As[m][k4i] = w4;
        }

        // prefetch next K tile into the other buffer, then retire current one
        const int kn = kk + 32;
        if (kn < 2 * CDIM) {
            const float* srcn = (kn < CDIM) ? (fB + kn * 1024) : (lB + (kn - CDIM) * 1024);
#pragma unroll
            for (int i = 0; i < 4; ++i) {
                const int idx = tid + i * 256;
                const int k = idx >> 5, n4 = (idx & 31) * 4;
                async_copy_b128(&Raw[buf ^ 1][k][n4], &srcn[k * 1024 + n4]);
            }
            asm volatile("s_wait_asynccnt 4" ::: "memory");  // current tile landed
        } else {
            asm volatile("s_wait_asynccnt 0" ::: "memory");
        }
        __syncthreads();

        const float (*Bmat)[136] = Raw[buf];
        if (kk >= CDIM) {
            // lateral half: Bs[k][n] = dot(A[w,:], l_row)
            for (int idx = tid; idx < 4096; idx += 256) {
                const int k = idx >> 7, n = idx & 127;
                const int nh = n >> 5, w = n & 31;
                float s = 0.f;
#pragma unroll
                for (int m = 0; m < 32; ++m) s = fmaf(Am[w][m], Raw[buf][k][nh * 32 + m], s);
                Bs[k][n] = s;
            }
            __syncthreads();
            Bmat = Bs;
        }

        // WMMA over the 32-deep K tile, 4 at a time: 8 WMMAs per k-step.
#pragma unroll
        for (int k4 = 0; k4 < 32; k4 += 4) {
            const int kA = k4 + (half ? 2 : 0);  // lanes 0-15: K=k4,k4+1; 16-31: k4+2,k4+3
            v2f a[2], bf[4];
            const int r0 = wm * 32 + l16;
            a[0].x = As[r0][kA];      a[0].y = As[r0][kA + 1];
            a[1].x = As[r0 + 16][kA]; a[1].y = As[r0 + 16][kA + 1];
#pragma unroll
            for (int jt = 0; jt < 4; ++jt) {
                const int cj = wn * 64 + jt * 16 + l16;
                bf[jt].x = Bmat[kA][cj];
                bf[jt].y = Bmat[kA + 1][cj];
            }
#pragma unroll
            for (int it = 0; it < 2; ++it)
#pragma unroll
                for (int jt = 0; jt < 4; ++jt)
                    acc[it][jt] = wmma_f32(a[it], bf[jt], acc[it][jt]);
        }
        __syncthreads();
    }

    // store: C/D layout -> flat column n within the 128-wide (4 h-rows) span
    float* outB = out + (long)b * CDIM * 1024 + h0 * 32;
#pragma unroll
    for (int it = 0; it < 2; ++it)
#pragma unroll
        for (int jt = 0; jt < 4; ++jt)
#pragma unroll
            for (int v = 0; v < 8; ++v) {
                const int o = mb * 128 + wm * 32 + it * 16 + v + (half ? 8 : 0);
                const int n = wn * 64 + jt * 16 + l16;
                outB[o * 1024 + n] = acc[it][jt][v];
            }
}

// ---------------------------------------------------------------------------
extern "C" void kernel_launch(void* const* d_in, const int* in_sizes, int n_in,
                              void* d_out, int out_size, void* d_ws, size_t ws_size,
                              hipStream_t stream) {
    const float* f  = (const float*)d_in[0];  // frontal_features [B,C,H,W]
    const float* l  = (const float*)d_in[1];  // lateral_features [B,C,H,W]
    const float* Wf = (const float*)d_in[2];  // w_frontal [C, 2C]
    float* out = (float*)d_out;               // [out | lateral_copy]
    float* Aws = (float*)d_ws;                // 1024 floats (softmax matrix)

    (void)in_sizes; (void)n_in; (void)out_size; (void)ws_size;

    // Gram partials are staged in d_out[0 .. 256*1024): that region is fully
    // overwritten by fused_conv_kernel afterwards, so no extra workspace.
    float* partials = out;

    corr_kernel<<<dim3(CORR_BLOCKS), dim3(256), 0, stream>>>(f, l, partials, out + NFLAT);
    softmax_kernel<<<dim3(1), dim3(32, 32), 0, stream>>>(partials, Aws);
    fused_conv_kernel<<<dim3(8, 8, 32), dim3(256), 0, stream>>>(f, l, Wf, Aws, out);
}